// RoiAlign_46084999086522
// MI455X (gfx1250) — compile-verified
//
#include <hip/hip_runtime.h>
#include <stdint.h>

typedef __attribute__((ext_vector_type(2))) float v2f;
typedef __attribute__((ext_vector_type(8))) float v8f;

#define CROP    14
#define CH      256
#define PSTRIDE 272      // padded floats per pixel-slot row (bank-shift between K rows)
#define NSLOT   64       // 2 feature rows x 32 column slots

// Per-axis sample geometry, matching the reference _crop_and_resize corners():
//   cs = (c1n*(1-t) + c2n*t) * (s-1);  lo=floor(cs); frac=cs-lo;
//   lo_i=clip(lo,0,s-1); hi_i=clip(lo+1,0,s-1); valid = 0<=cs<=s-1
__device__ __forceinline__ void sample_axis(float c1n, float c2n, float sf, int j,
                                            int& lo_i, int& hi_i, float& fr, bool& valid) {
  float t  = (float)j / 13.0f;
  float cs = (c1n * (1.0f - t) + c2n * t) * (sf - 1.0f);
  float lo = floorf(cs);
  fr   = cs - lo;
  lo_i = (int)fminf(fmaxf(lo,        0.0f), sf - 1.0f);
  hi_i = (int)fminf(fmaxf(lo + 1.0f, 0.0f), sf - 1.0f);
  valid = (cs >= 0.0f) && (cs <= sf - 1.0f);
}

// CDNA5 async global->LDS copy (ASYNCcnt path); VDST VGPR carries LDS byte addr.
__device__ __forceinline__ void async_b128(uint32_t lds_byte, const void* g) {
  unsigned long long ga = (unsigned long long)g;
  asm volatile("global_load_async_to_lds_b128 %0, %1, off" :: "v"(lds_byte), "v"(ga) : "memory");
}
__device__ __forceinline__ void wait_async0() {
  asm volatile("s_wait_asynccnt 0x0" ::: "memory");
}

__global__ __launch_bounds__(256) void roi_align_wmma(
    const int* __restrict__ image_shape, const float* __restrict__ boxes,
    const float* __restrict__ f0, const float* __restrict__ f1, const float* __restrict__ f2,
    const float* __restrict__ f3, const float* __restrict__ f4, float* __restrict__ out)
{
  extern __shared__ float Plds[];              // NSLOT * PSTRIDE floats (~70 KB)

  const int i    = blockIdx.x;                 // output row 0..13
  const int b    = blockIdx.y;                 // box
  const int tid  = threadIdx.x;
  const int lane = tid & 31, wv = tid >> 5, hi = lane >> 4, jl = lane & 15;

  // ---- box geometry (all threads, uniform per block) ----
  float bx1 = boxes[b*4+0], by1 = boxes[b*4+1], bx2 = boxes[b*4+2], by2 = boxes[b*4+3];
  float size = sqrtf((bx2 - bx1) * (by2 - by1));
  float lvf  = fminf(fmaxf(floorf(1.0f + log2f(size * (1.0f/224.0f) + 1e-7f)), 0.0f), 4.0f);
  int   lv   = (int)lvf;
  const float* F = (lv==0)?f0:(lv==1)?f1:(lv==2)?f2:(lv==3)?f3:f4;
  int   s  = 256 >> lv;
  float sf = (float)s;
  float img_h = (float)image_shape[1];
  float img_w = (float)image_shape[2];
  float y1n =  by1/img_h * sf        / (sf-1.0f);
  float x1n =  bx1/img_w * sf        / (sf-1.0f);
  float y2n = (by2/img_h * sf - 1.0f) / (sf-1.0f);
  float x2n = (bx2/img_w * sf - 1.0f) / (sf-1.0f);

  int y0, y1c; float fy; bool vy;
  sample_axis(y1n, y2n, sf, i, y0, y1c, fy, vy);

  // ---- stage 64 slots x 256 ch into LDS via async copies ----
  {
    int slot = tid >> 2, q = tid & 3;          // 4 threads per slot
    int r = slot >> 5, xslot = slot & 31, jj = xslot >> 1, xx = xslot & 1;
    int col = 0;
    if (jj < CROP) {
      int x0g, x1g; float fxg; bool vxg;
      sample_axis(x1n, x2n, sf, jj, x0g, x1g, fxg, vxg);
      col = xx ? x1g : x0g;
    }
    int rowg = r ? y1c : y0;
    const char* gsrc = (const char*)(F + ((size_t)rowg * (size_t)s + (size_t)col) * CH);
    uint32_t lbase = (uint32_t)(size_t)&Plds[slot * PSTRIDE];
    #pragma unroll
    for (int it = 0; it < 16; ++it) {
      uint32_t off = (uint32_t)(it * 64 + q * 16);   // bytes; 4 lanes cover 64B contiguous
      async_b128(lbase + off, (const void*)(gsrc + off));
    }
  }
  wait_async0();
  __syncthreads();

  // ---- per-lane weight column (W[:,jl]); mask folded into weights ----
  int x0l, x1l; float fxl; bool vxl;
  sample_axis(x1n, x2n, sf, jl, x0l, x1l, fxl, vxl);
  float msk = (vxl && jl < CROP) ? 1.0f : 0.0f;
  float wy0 = vy ? (1.0f - fy) : 0.0f;
  float wy1 = vy ? fy          : 0.0f;
  float wx0 = msk * (1.0f - fxl);
  float wx1 = msk * fxl;

  float* ob = out + (((size_t)b * CROP + (size_t)i) * CROP) * CH;

  // ---- out(14x256) = W(14x64) x P(64x256) as 16 chained f32 16x16x4 WMMAs ----
  #pragma unroll
  for (int tt = 0; tt < 2; ++tt) {
    int t  = wv + tt * 8;                      // channel tile 0..15
    int ch = t * 16 + jl;                      // this lane's channel column (N)
    v8f acc = {0.f,0.f,0.f,0.f,0.f,0.f,0.f,0.f};
    #pragma unroll
    for (int kk = 0; kk < 16; ++kk) {
      int R0 = 4*kk + 2*hi;                    // this lane-half's two K rows: R0, R0+1
      // W[k][j] nonzero iff slot (k&31) belongs to column j: (k&31)>>1 == j
      bool  hit = (((R0 & 31) >> 1) == jl);
      float wyr = (R0 >> 5) ? wy1 : wy0;       // which feature row (y0 vs y1)
      v2f a;  a.x  = hit ? wyr * wx0 : 0.0f;   // k even -> x0 corner
              a.y  = hit ? wyr * wx1 : 0.0f;   // k odd  -> x1 corner
      v2f bb; bb.x = Plds[R0      * PSTRIDE + ch];
              bb.y = Plds[(R0+1)  * PSTRIDE + ch];
      acc = __builtin_amdgcn_wmma_f32_16x16x4_f32(false, a, false, bb,
                                                  (short)0, acc, false, false);
    }
    // D layout: lane&15 = N (channel), VGPR v = M row j (+8 for upper lane half)
    #pragma unroll
    for (int v = 0; v < 8; ++v) {
      int jo = v + 8 * hi;
      if (jo < CROP) ob[(size_t)jo * CH + ch] = acc[v];
    }
  }
}

extern "C" void kernel_launch(void* const* d_in, const int* in_sizes, int n_in,
                              void* d_out, int out_size, void* d_ws, size_t ws_size,
                              hipStream_t stream) {
  const int*   image_shape = (const int*)d_in[0];
  const float* boxes       = (const float*)d_in[1];
  // d_in[2] = scores (unused by the reference output)
  const float* f0 = (const float*)d_in[3];
  const float* f1 = (const float*)d_in[4];
  const float* f2 = (const float*)d_in[5];
  const float* f3 = (const float*)d_in[6];
  const float* f4 = (const float*)d_in[7];
  float* out = (float*)d_out;

  int nbox = in_sizes[1] / 4;                  // (1, N, 4) -> N
  dim3 grid(CROP, nbox);                       // one WG per (output row, box)
  size_t lds = (size_t)NSLOT * PSTRIDE * sizeof(float);
  roi_align_wmma<<<grid, 256, lds, stream>>>(image_shape, boxes, f0, f1, f2, f3, f4, out);
}